// QuantumCorrelationTracker_89154931130743
// MI455X (gfx1250) — compile-verified
//
#include <hip/hip_runtime.h>

// CDNA5 / gfx1250, wave32.
// Quantum mutual information for rank-1 density matrices collapses to:
//   S(x) = -lam*log(lam) + (D-1)*(-1e-12*log(1e-12)),  lam = sumsq(x)/max(sumsq(x),1e-24)
// sumsq per row is computed as the diagonal of G = X X^T via V_WMMA_F32_16X16X4_F32.

typedef float v2f __attribute__((ext_vector_type(2)));
typedef float v8f __attribute__((ext_vector_type(8)));

#define D_DIM 256

__device__ __forceinline__ float entropy_from_q(float q) {
    // lam = sumsq / max(||x||,1e-12)^2 ; max(||x||,eps)^2 == max(sumsq, eps^2) for sumsq>=0
    float lam = q / fmaxf(q, 1e-24f);
    lam = fmaxf(lam, 1e-12f);                       // clip(ev, 1e-12)
    // 255 clamped zero-eigenvalues: each contributes -1e-12*log(1e-12)
    const float TAIL = 255.0f * 2.7631021115928547e-11f;   // ~7.0459e-9
    return -lam * __logf(lam) + TAIL;
}

__global__ __launch_bounds__(32)
void qmi_rank1_wmma_kernel(const float* __restrict__ s1,
                           const float* __restrict__ s2,
                           const float* __restrict__ s12,
                           float* __restrict__ out) {
    const int l   = threadIdx.x;        // 0..31, one wave
    const int row = l & 15;             // M (and N) index within 16-row tile
    const int h   = l >> 4;             // lane half selects K sub-pair
    const long base = (long)(blockIdx.x * 16 + row) * D_DIM + 2 * h;

    const float* p1 = s1  + base;
    const float* p2 = s2  + base;
    const float* p3 = s12 + base;

    v8f c1 = {}, c2 = {}, c3 = {};

    // Walk K = 256 in steps of 4 (16x16x4 f32 WMMA). A and B per-lane operand
    // values coincide for X*X^T under the ISA A/B 32-bit layouts.
    #pragma unroll 4
    for (int j = 0; j < D_DIM / 4; ++j) {
        v2f a1 = *(const v2f*)(p1 + 4 * j);
        v2f a2 = *(const v2f*)(p2 + 4 * j);
        v2f a3 = *(const v2f*)(p3 + 4 * j);
        c1 = __builtin_amdgcn_wmma_f32_16x16x4_f32(false, a1, false, a1,
                                                   (short)0, c1, false, false);
        c2 = __builtin_amdgcn_wmma_f32_16x16x4_f32(false, a2, false, a2,
                                                   (short)0, c2, false, false);
        c3 = __builtin_amdgcn_wmma_f32_16x16x4_f32(false, a3, false, a3,
                                                   (short)0, c3, false, false);
    }

    // Diagonal extraction. C/D layout (32-bit, 16x16):
    //   lanes 0-15 : VGPR r -> (M=r,   N=lane)
    //   lanes 16-31: VGPR r -> (M=r+8, N=lane-16)
    // diag(m): m<8 -> VGPR m at lane m ; m>=8 -> VGPR m-8 at lane m+16.
    // So per-lane select c[l&7]; diagonals live on lanes 0-7 and 24-31.
    const int r = l & 7;
    float q1 = c1[0], q2 = c2[0], q3 = c3[0];
    #pragma unroll
    for (int k = 1; k < 8; ++k) {
        q1 = (r == k) ? c1[k] : q1;
        q2 = (r == k) ? c2[k] : q2;
        q3 = (r == k) ? c3[k] : q3;
    }

    const bool valid = (l < 8) || (l >= 24);
    const int  m     = (l < 8) ? l : (l - 16);

    if (valid) {
        float s = entropy_from_q(q1) + entropy_from_q(q2) - entropy_from_q(q3);
        out[blockIdx.x * 16 + m] = s;
    }
}

extern "C" void kernel_launch(void* const* d_in, const int* in_sizes, int n_in,
                              void* d_out, int out_size, void* d_ws, size_t ws_size,
                              hipStream_t stream) {
    const float* s1  = (const float*)d_in[0];
    const float* s2  = (const float*)d_in[1];
    const float* s12 = (const float*)d_in[2];
    float* out = (float*)d_out;

    const int B     = in_sizes[0] / D_DIM;   // 256
    const int tiles = B / 16;                // 16 blocks, 1 wave each

    qmi_rank1_wmma_kernel<<<tiles, 32, 0, stream>>>(s1, s2, s12, out);
}